// IDOModel_77910706749489
// MI455X (gfx1250) — compile-verified
//
#include <hip/hip_runtime.h>

// ---------------------------------------------------------------------------
// AdaCoF (T=2, B=4, C=3, H=W=256, KS=5) for MI455X / gfx1250.
// Bandwidth-bound streaming kernel: w/alpha/beta tiles staged to LDS with the
// Tensor Data Mover (tensor_load_to_lds, TENSORcnt), frames gathered from L2.
// 128-column tiles -> 76.8 KB LDS/block -> up to 4 blocks (16 waves) per WGP.
// ---------------------------------------------------------------------------

#define KS    5
#define K2    25
#define PADR  2            // (KS-1)*DIL/2
#define TT    2
#define BB    4
#define CC    3
#define HH    256
#define WW    256
#define TILEW 128          // columns per block
#define HWSZ  (HH * WW)

typedef __attribute__((ext_vector_type(4))) unsigned int u32x4;
typedef __attribute__((ext_vector_type(8))) int          i32x8;
typedef __attribute__((ext_vector_type(4))) int          i32x4;

__device__ __forceinline__ int rfl(int x) {
  return __builtin_amdgcn_readfirstlane(x);
}

// Issue one 2-D TDM load: tile_d1 rows of tile_d0 fp32 elements, rows strided
// by d0_stride elements in global memory, packed densely into LDS at lds_off.
__device__ __forceinline__ void tdm_load_2d(const float* gptr,
                                            unsigned lds_off,
                                            unsigned tensor_d0, unsigned tensor_d1,
                                            unsigned tile_d0,   unsigned tile_d1,
                                            unsigned long long d0_stride_elems)
{
  unsigned long long ga = (unsigned long long)(uintptr_t)gptr;

  u32x4 g0;
  g0.x = (unsigned)rfl(1);                                   // count=1, user D#
  g0.y = (unsigned)rfl((int)lds_off);                        // lds_addr [63:32]
  g0.z = (unsigned)rfl((int)(ga & 0xffffffffu));             // global_addr[31:0]
  g0.w = (unsigned)rfl((int)(((ga >> 32) & 0x01ffffffu) | (2u << 30))); // [56:32] | type=2

  i32x8 g1;
  g1[0] = rfl((int)(2u << 16));                              // data_size = 4B; mask=0
  g1[1] = rfl((int)((tensor_d0 & 0xffffu) << 16));           // tensor_dim0[15:0] @ bit48
  g1[2] = rfl((int)(((tensor_d0 >> 16) & 0xffffu) |
                    ((tensor_d1 & 0xffffu) << 16)));         // dim0[31:16] | dim1[15:0]
  g1[3] = rfl((int)(((tensor_d1 >> 16) & 0xffffu) |
                    ((tile_d0 & 0xffffu) << 16)));           // dim1[31:16] | tile_dim0
  g1[4] = rfl((int)(tile_d1 & 0xffffu));                     // tile_dim1; tile_dim2=0
  g1[5] = rfl((int)(d0_stride_elems & 0xffffffffull));       // dim0_stride[31:0]
  g1[6] = rfl((int)((d0_stride_elems >> 32) & 0xffffull));   // dim0_stride[47:32]; dim1_stride=0
  g1[7] = 0;

  i32x4 z4 = {0, 0, 0, 0};                                   // 2-D tensor: groups 2/3 unused
  i32x8 z8 = {0, 0, 0, 0, 0, 0, 0, 0};
  __builtin_amdgcn_tensor_load_to_lds(g0, g1, z4, z4, z8, 0);
}

__global__ __launch_bounds__(TILEW)
void adacof_fused_kernel(const float* __restrict__ frames,   // [T,B,C,H,W]
                         const float* __restrict__ weights,  // [B,T,K2,H,W]
                         const float* __restrict__ alphas,   // [B,T,K2,H,W]
                         const float* __restrict__ betas,    // [B,T,K2,H,W]
                         const float* __restrict__ occ,      // [B,T,H,W]
                         float* __restrict__ out)            // [B,C,H,W]
{
  __shared__ float s_w[TT][K2][TILEW];
  __shared__ float s_a[TT][K2][TILEW];
  __shared__ float s_b[TT][K2][TILEW];

  const int jt   = threadIdx.x;                  // column within tile
  const int half = blockIdx.x & 1;               // which half-row
  const int i    = (blockIdx.x >> 1) & (HH - 1); // row
  const int b    = blockIdx.x >> 9;              // batch (H*2 = 512 blocks per b)
  const int j    = half * TILEW + jt;            // global column

  // ---- Wave 0: stage the six k-strided tiles into LDS via the TDM ----
  if (threadIdx.x < 32) {
    for (int t = 0; t < TT; ++t) {
      const long long base = ((long long)(b * TT + t)) * K2 * HWSZ
                           + (long long)i * WW + half * TILEW;
      tdm_load_2d(weights + base, (unsigned)(uintptr_t)&s_w[t][0][0],
                  TILEW, K2, TILEW, K2, (unsigned long long)HWSZ);
      tdm_load_2d(alphas + base, (unsigned)(uintptr_t)&s_a[t][0][0],
                  TILEW, K2, TILEW, K2, (unsigned long long)HWSZ);
      tdm_load_2d(betas + base, (unsigned)(uintptr_t)&s_b[t][0][0],
                  TILEW, K2, TILEW, K2, (unsigned long long)HWSZ);
    }
    __builtin_amdgcn_s_wait_tensorcnt(0);
  }
  __syncthreads();

  // ---- Occlusion softmax over T=2 ----
  const int pix = i * WW + j;
  const float l0 = occ[(b * TT + 0) * HWSZ + pix];
  const float l1 = occ[(b * TT + 1) * HWSZ + pix];
  const float lm = fmaxf(l0, l1);
  const float e0 = __expf(l0 - lm);
  const float e1 = __expf(l1 - lm);
  const float oinv = 1.0f / (e0 + e1);
  const float occw0 = e0 * oinv;
  const float occw1 = e1 * oinv;

  float acc0 = 0.f, acc1 = 0.f, acc2 = 0.f;
  const float fi = (float)i;
  const float fj = (float)j;
  const int Hp = HH + 2 * PADR;
  const int Wp = WW + 2 * PADR;

  for (int t = 0; t < TT; ++t) {
    // softmax-max over the 25 kernel taps (LDS reads)
    float wmax = -3.4e38f;
    #pragma unroll
    for (int k = 0; k < K2; ++k) wmax = fmaxf(wmax, s_w[t][k][jt]);

    float wsum = 0.f;
    float c0 = 0.f, c1 = 0.f, c2 = 0.f;
    const float* fbase = frames + ((long long)(t * BB + b)) * CC * HWSZ;
    const float* f0 = fbase;
    const float* f1 = fbase + HWSZ;
    const float* f2 = fbase + 2 * HWSZ;

    #pragma unroll 5
    for (int k = 0; k < K2; ++k) {
      const float wk = __expf(s_w[t][k][jt] - wmax);
      wsum += wk;

      const float dy = (float)(k / KS);
      const float dx = (float)(k % KS);
      float y = s_a[t][k][jt] + dy + fi;
      float x = s_b[t][k][jt] + dx + fj;
      y = fminf(fmaxf(y, 0.f), (float)(Hp - 1));
      x = fminf(fmaxf(x, 0.f), (float)(Wp - 1));

      int y0 = (int)floorf(y);
      int x0 = (int)floorf(x);
      if (y0 > Hp - 2) y0 = Hp - 2;
      if (x0 > Wp - 2) x0 = Wp - 2;
      const float fy = y - (float)y0;
      const float fx = x - (float)x0;

      // edge-replication pad folded into clamped unpadded-frame coords
      int yA = y0 - PADR, yB = y0 + 1 - PADR;
      int xA = x0 - PADR, xB = x0 + 1 - PADR;
      yA = min(max(yA, 0), HH - 1);
      yB = min(max(yB, 0), HH - 1);
      xA = min(max(xA, 0), WW - 1);
      xB = min(max(xB, 0), WW - 1);

      const float w00 = (1.f - fy) * (1.f - fx);
      const float w01 = (1.f - fy) * fx;
      const float w10 = fy * (1.f - fx);
      const float w11 = fy * fx;
      const int rA = yA * WW, rB = yB * WW;

      const float v0 = w00 * f0[rA + xA] + w01 * f0[rA + xB]
                     + w10 * f0[rB + xA] + w11 * f0[rB + xB];
      const float v1 = w00 * f1[rA + xA] + w01 * f1[rA + xB]
                     + w10 * f1[rB + xA] + w11 * f1[rB + xB];
      const float v2 = w00 * f2[rA + xA] + w01 * f2[rA + xB]
                     + w10 * f2[rB + xA] + w11 * f2[rB + xB];

      c0 += wk * v0;
      c1 += wk * v1;
      c2 += wk * v2;
    }

    const float s = (t == 0 ? occw0 : occw1) / wsum;
    acc0 += s * c0;
    acc1 += s * c1;
    acc2 += s * c2;
  }

  float* ob = out + ((long long)b * CC) * HWSZ + pix;
  ob[0]        = acc0;
  ob[HWSZ]     = acc1;
  ob[2 * HWSZ] = acc2;
}

extern "C" void kernel_launch(void* const* d_in, const int* in_sizes, int n_in,
                              void* d_out, int out_size, void* d_ws, size_t ws_size,
                              hipStream_t stream) {
  (void)in_sizes; (void)n_in; (void)out_size; (void)d_ws; (void)ws_size;
  const float* frames  = (const float*)d_in[0];
  const float* weights = (const float*)d_in[1];
  const float* alphas  = (const float*)d_in[2];
  const float* betas   = (const float*)d_in[3];
  const float* occ     = (const float*)d_in[4];
  float* out = (float*)d_out;

  dim3 grid(BB * HH * 2);
  dim3 block(TILEW);
  hipLaunchKernelGGL(adacof_fused_kernel, grid, block, 0, stream,
                     frames, weights, alphas, betas, occ, out);
}